// GraphLearning_32220844655187
// MI455X (gfx1250) — compile-verified
//
#include <hip/hip_runtime.h>

typedef __attribute__((ext_vector_type(2))) float v2f;
typedef __attribute__((ext_vector_type(8))) float v8f;

static constexpr int kB  = 8;
static constexpr int kF  = 128;
static constexpr int kN  = 2048;
static constexpr int kNT = kN / 16;    // 128 tiles per dimension
static constexpr int kTJ = 8;          // j-tiles per wave (16x128 strip)
static constexpr int kJobsPerRow = kNT / kTJ;   // 16

// ---------------------------------------------------------------------------
// Kernel 1: projections a[b,n] = X[b,:,n].w1, c[b,n] = X[b,:,n].w2
// One thread per (b,n); loads over f are coalesced across n. W reads are
// wave-uniform -> scalar loads. Writes 2 * 64 KB into workspace.
// ---------------------------------------------------------------------------
__global__ __launch_bounds__(256) void proj_kernel(
    const float* __restrict__ X, const float* __restrict__ W,
    float* __restrict__ a, float* __restrict__ c)
{
    int idx = blockIdx.x * 256 + threadIdx.x;   // b*kN + n
    int b = idx >> 11;                          // / 2048
    int n = idx & (kN - 1);
    const float* xp = X + (size_t)b * kF * kN + n;
    float sa = 0.0f, sc = 0.0f;
#pragma unroll 4
    for (int f = 0; f < kF; ++f) {
        float x = xp[(size_t)f * kN];
        sa = __builtin_fmaf(x, W[f], sa);
        sc = __builtin_fmaf(x, W[kF + f], sc);
    }
    a[idx] = sa;
    c[idx] = sc;
}

// ---------------------------------------------------------------------------
// Kernel 2: one wave per 16x128 output strip (1 i-tile x 8 j-tiles).
//   out[b,i,j] = (i==j) ? 0 : sigmoid(a[b,min] + c[b,max] + bias)
// The 16-row broadcast uses V_WMMA_F32_16X16X4_F32 once per strip:
//   A: 16x4, only K=0 populated (VGPR0 lanes 0-15 = rowval, per ISA layout)
//   B: 4x16, only row K=0 = 1.0 (other K rows zeroed -> layout-robust)
//   => D[m][n] = rowval[m]; column value + bias added per-lane in epilogue.
// wa/wc depend only on ti, so both WMMA results are reused across all 8
// j-tiles of the strip. Sigmoid uses fast v_rcp_f32 (no IEEE div chain).
// ---------------------------------------------------------------------------
__global__ __launch_bounds__(256) void score_kernel(
    const float* __restrict__ a, const float* __restrict__ c,
    const float* __restrict__ bptr, float* __restrict__ out)
{
    const int lane = threadIdx.x & 31;
    const int wave = (blockIdx.x * 256 + threadIdx.x) >> 5;
    const int jobsPerBatch = kNT * kJobsPerRow;         // 2048
    const int b   = wave >> 11;                         // / 2048
    const int rem = wave & (jobsPerBatch - 1);
    const int ti  = rem >> 4;                           // / 16
    const int tjb = (rem & (kJobsPerRow - 1)) * kTJ;    // first j-tile

    const float bias = bptr[0];
    const int  l15 = lane & 15;
    const bool lo  = lane < 16;

    const float* ab = a + (size_t)b * kN;
    const float* cb = c + (size_t)b * kN;

    // Row operands (only lanes 0-15 feed the A K=0 slice).
    float a_row = ab[ti * 16 + l15];
    float c_row = cb[ti * 16 + l15];

    v2f A1; A1.x = lo ? a_row : 0.0f; A1.y = 0.0f;   // rows = a (i<j path)
    v2f A2; A2.x = lo ? c_row : 0.0f; A2.y = 0.0f;   // rows = c (i>j path)
    v2f Bv; Bv.x = lo ? 1.0f  : 0.0f; Bv.y = 0.0f;   // ones in K=0 row only

    v8f Z = {0.f, 0.f, 0.f, 0.f, 0.f, 0.f, 0.f, 0.f};
    v8f wa = __builtin_amdgcn_wmma_f32_16x16x4_f32(
        false, A1, false, Bv, (short)0, Z, false, false);
    v8f wc = __builtin_amdgcn_wmma_f32_16x16x4_f32(
        false, A2, false, Bv, (short)0, Z, false, false);

    const int rowbase = ti * 16 + (lo ? 0 : 8);   // C/D layout: M = r (+8 hi half)
    float* op = out + (size_t)b * kN * kN;

    for (int tj = tjb; tj < tjb + kTJ; ++tj) {
        // Column values: per-lane scalars (C/D layout: N = lane & 15).
        float c_colb = cb[tj * 16 + l15] + bias;
        float a_colb = ab[tj * 16 + l15] + bias;
        int gj = tj * 16 + l15;

#pragma unroll
        for (int r = 0; r < 8; ++r) {
            int gi = rowbase + r;
            float s = (gi < gj) ? (wa[r] + c_colb) : (wc[r] + a_colb);
            float e = __expf(-s);                       // v_mul + v_exp_f32
            float v = __builtin_amdgcn_rcpf(1.0f + e);  // fast v_rcp_f32
            if (gi == gj) v = 0.0f;
            // Lanes 0-15 / 16-31 each write one contiguous 64B row segment.
            op[(size_t)gi * kN + gj] = v;
        }
    }
}

extern "C" void kernel_launch(void* const* d_in, const int* in_sizes, int n_in,
                              void* d_out, int out_size, void* d_ws, size_t ws_size,
                              hipStream_t stream) {
    const float* X  = (const float*)d_in[0];   // node_features (8,128,2048)
    const float* W  = (const float*)d_in[1];   // (1, 256)
    const float* bb = (const float*)d_in[2];   // (1,)
    float* out = (float*)d_out;                // (8, 2048, 2048)

    float* a = (float*)d_ws;                   // 8*2048 f32 = 64 KB
    float* c = a + kB * kN;                    // next 64 KB

    // Stage 1: projections. 16384 threads.
    proj_kernel<<<(kB * kN) / 256, 256, 0, stream>>>(X, W, a, c);

    // Stage 2: 8 * 128 * 16 = 16384 strip-waves, 8 waves/block.
    score_kernel<<<(kB * kNT * kJobsPerRow * 32) / 256, 256, 0, stream>>>(
        a, c, bb, out);
}